// StudentT_KSDLoss_78932908965975
// MI455X (gfx1250) — compile-verified
//
#include <hip/hip_runtime.h>
#include <math.h>

typedef __attribute__((ext_vector_type(2))) float v2f;
typedef __attribute__((ext_vector_type(8))) float v8f;

constexpr int kN = 4096;
constexpr int kD = 256;

struct Ctrl {
    unsigned prefix;   // radix-select prefix (high bits fixed so far)
    unsigned kth;      // remaining 0-based rank within current prefix class
    float    alpha;    // 1/(median + 1e-6), written after pass 3
    unsigned pad;
};

// ---------------------------------------------------------------------------
// Kernel 1: row norms + Student-t score coefficient  c_i = -(nu+d)/(nu+|z_i|^2)
// ---------------------------------------------------------------------------
__global__ void norms_kernel(const float* __restrict__ z,
                             float* __restrict__ ns,
                             float* __restrict__ cc) {
    __shared__ float red[kD];
    const int i = blockIdx.x;
    const float v = z[i * kD + threadIdx.x];
    red[threadIdx.x] = v * v;
    __syncthreads();
    for (int s = kD / 2; s > 0; s >>= 1) {
        if ((int)threadIdx.x < s) red[threadIdx.x] += red[threadIdx.x + s];
        __syncthreads();
    }
    if (threadIdx.x == 0) {
        const float n2 = red[0];
        ns[i] = n2;
        cc[i] = -(3.0f + (float)kD) / (3.0f + n2);   // nu=3, sigma=1
    }
}

// ---------------------------------------------------------------------------
// Kernel 2: init radix-select state
// ---------------------------------------------------------------------------
__global__ void init_kernel(unsigned* __restrict__ bins, Ctrl* __restrict__ ctrl) {
    bins[threadIdx.x] = 0u;
    if (threadIdx.x == 0) {
        ctrl->prefix = 0u;
        ctrl->kth    = ((unsigned)kN * (unsigned)kN - 1u) / 2u;  // 8388607
        ctrl->alpha  = 0.0f;
    }
}

// ---------------------------------------------------------------------------
// In-register 32x32 Gram tile via V_WMMA_F32_16X16X4_F32 (fp32-exact GEMM).
// Per ISA layout: lane l (r=l&15, h=l>>4): A frag = z[i0+r, k+2h .. k+2h+1],
// B frag = z[j0+r, k+2h .. k+2h+1].  C/D: elem g of lane l -> (row g+8h, col r).
// ---------------------------------------------------------------------------
__device__ __forceinline__ void gram_tile32(const float* __restrict__ z,
                                            int i0, int j0, int r, int h,
                                            v8f acc[2][2]) {
    const float* a0p = z + (i0 + r)      * kD + 2 * h;
    const float* a1p = z + (i0 + 16 + r) * kD + 2 * h;
    const float* b0p = z + (j0 + r)      * kD + 2 * h;
    const float* b1p = z + (j0 + 16 + r) * kD + 2 * h;
#pragma unroll 8
    for (int k = 0; k < kD; k += 4) {
        v2f a0 = *(const v2f*)(a0p + k);
        v2f a1 = *(const v2f*)(a1p + k);
        v2f b0 = *(const v2f*)(b0p + k);
        v2f b1 = *(const v2f*)(b1p + k);
        acc[0][0] = __builtin_amdgcn_wmma_f32_16x16x4_f32(false, a0, false, b0,
                        (short)0, acc[0][0], false, false);
        acc[0][1] = __builtin_amdgcn_wmma_f32_16x16x4_f32(false, a0, false, b1,
                        (short)0, acc[0][1], false, false);
        acc[1][0] = __builtin_amdgcn_wmma_f32_16x16x4_f32(false, a1, false, b0,
                        (short)0, acc[1][0], false, false);
        acc[1][1] = __builtin_amdgcn_wmma_f32_16x16x4_f32(false, a1, false, b1,
                        (short)0, acc[1][1], false, false);
    }
}

// ---------------------------------------------------------------------------
// Kernel 3: radix-select histogram pass over dist^2 (recomputed via WMMA).
// Block = 128 threads = 4 waves; block tile = 64x64; wave tile = 32x32.
// ---------------------------------------------------------------------------
__global__ void hist_kernel(const float* __restrict__ z,
                            const float* __restrict__ ns,
                            unsigned* __restrict__ bins,
                            const Ctrl* __restrict__ ctrl,
                            int pass) {
    __shared__ unsigned lb[256];
    const int t = threadIdx.x;
    lb[t] = 0u; lb[t + 128] = 0u;

    const int w = t >> 5, lane = t & 31;
    const int r = lane & 15, h = lane >> 4;
    const int i0 = blockIdx.y * 64 + (w & 1) * 32;
    const int j0 = blockIdx.x * 64 + (w >> 1) * 32;

    v8f acc[2][2] = {};
    gram_tile32(z, i0, j0, r, h, acc);
    __syncthreads();

    const int shift = 24 - 8 * pass;
    const unsigned long long pfx_hi =
        ((unsigned long long)ctrl->prefix) >> (shift + 8);

#pragma unroll
    for (int mi = 0; mi < 2; ++mi) {
#pragma unroll
        for (int nj = 0; nj < 2; ++nj) {
            const int col = j0 + nj * 16 + r;
            const float nsj = ns[col];
#pragma unroll
            for (int g = 0; g < 8; ++g) {
                const int row = i0 + mi * 16 + g + 8 * h;
                const float d =
                    fmaxf(ns[row] + nsj - 2.0f * acc[mi][nj][g], 0.0f);
                const unsigned bits = __float_as_uint(d);
                if (pass == 0 ||
                    ((((unsigned long long)bits) >> (shift + 8)) == pfx_hi)) {
                    atomicAdd(&lb[(bits >> shift) & 255u], 1u);
                }
            }
        }
    }
    __syncthreads();
    if (lb[t])       atomicAdd(&bins[t],       lb[t]);
    if (lb[t + 128]) atomicAdd(&bins[t + 128], lb[t + 128]);
}

// ---------------------------------------------------------------------------
// Kernel 4: pick the bucket containing the k-th value; refine prefix; clear bins.
// After pass 3 the full 32-bit pattern of the median is known -> alpha.
// ---------------------------------------------------------------------------
__global__ void select_kernel(unsigned* __restrict__ bins,
                              Ctrl* __restrict__ ctrl, int pass) {
    if (threadIdx.x == 0) {
        const unsigned k = ctrl->kth;
        unsigned cum = 0u, b = 255u;
        for (int i = 0; i < 256; ++i) {
            const unsigned cnt = bins[i];
            if (cum + cnt > k) { b = (unsigned)i; break; }
            cum += cnt;
        }
        ctrl->kth = k - cum;
        const int shift = 24 - 8 * pass;
        const unsigned prefix = ctrl->prefix | (b << shift);
        ctrl->prefix = prefix;
        if (pass == 3) {
            const float median = __uint_as_float(prefix);
            ctrl->alpha = 1.0f / (median + 1e-6f);
        }
    }
    __syncthreads();
    bins[threadIdx.x] = 0u;   // blockDim == 256: ready for next pass
}

// ---------------------------------------------------------------------------
// Kernel 5: fused Gram-recompute + Stein-kernel evaluation + block reduction.
//   base = 1 + a*d2 ; K = base^-1/2 ; grad = -a*base^-3/2 (beta = 1/2)
//   term_a = c_i c_j G K ; term_b = -grad*c_i*(ns_i - G) ; term_c = grad*c_j*(G - ns_j)
//   lap    = grad*(d - 3a*d2/base)
// ---------------------------------------------------------------------------
__global__ void reduce_kernel(const float* __restrict__ z,
                              const float* __restrict__ ns,
                              const float* __restrict__ cc,
                              const Ctrl* __restrict__ ctrl,
                              float* __restrict__ partials) {
    __shared__ float red[128];
    const int t = threadIdx.x;
    const int w = t >> 5, lane = t & 31;
    const int r = lane & 15, h = lane >> 4;
    const int i0 = blockIdx.y * 64 + (w & 1) * 32;
    const int j0 = blockIdx.x * 64 + (w >> 1) * 32;

    v8f acc[2][2] = {};
    gram_tile32(z, i0, j0, r, h, acc);

    const float alpha = ctrl->alpha;
    float sum = 0.0f;
#pragma unroll
    for (int mi = 0; mi < 2; ++mi) {
#pragma unroll
        for (int nj = 0; nj < 2; ++nj) {
            const int col = j0 + nj * 16 + r;
            const float nsj = ns[col];
            const float cj  = cc[col];
#pragma unroll
            for (int g = 0; g < 8; ++g) {
                const int row = i0 + mi * 16 + g + 8 * h;
                if (row == col) continue;            // exclude diagonal
                const float gm   = acc[mi][nj][g];
                const float nsi  = ns[row];
                const float ci   = cc[row];
                const float d2   = fmaxf(nsi + nsj - 2.0f * gm, 0.0f);
                const float base = 1.0f + alpha * d2;
                const float invb = 1.0f / base;
                const float Kv   = 1.0f / sqrtf(base);      // base^-1/2
                const float grad = -alpha * Kv * invb;      // -2*a*beta*base^-3/2
                const float ta   = ci * cj * gm * Kv;
                const float tb   = -grad * ci * (nsi - gm);
                const float tc   =  grad * cj * (gm - nsj);
                const float lap  = grad * ((float)kD - 3.0f * alpha * d2 * invb);
                sum += ta + tb + tc + lap;
            }
        }
    }
    red[t] = sum;
    __syncthreads();
    for (int s = 64; s > 0; s >>= 1) {
        if (t < s) red[t] += red[t + s];
        __syncthreads();
    }
    if (t == 0) partials[blockIdx.y * 64 + blockIdx.x] = red[0];
}

// ---------------------------------------------------------------------------
// Kernel 6: final deterministic sum of 4096 block partials (double accum).
// ---------------------------------------------------------------------------
__global__ void final_kernel(const float* __restrict__ partials,
                             float* __restrict__ out) {
    __shared__ double red[256];
    double s = 0.0;
    for (int i = threadIdx.x; i < 64 * 64; i += 256) s += (double)partials[i];
    red[threadIdx.x] = s;
    __syncthreads();
    for (int st = 128; st > 0; st >>= 1) {
        if ((int)threadIdx.x < st) red[threadIdx.x] += red[threadIdx.x + st];
        __syncthreads();
    }
    if (threadIdx.x == 0)
        out[0] = (float)(red[0] / ((double)kN * (double)(kN - 1)));
}

// ---------------------------------------------------------------------------
extern "C" void kernel_launch(void* const* d_in, const int* in_sizes, int n_in,
                              void* d_out, int out_size, void* d_ws, size_t ws_size,
                              hipStream_t stream) {
    (void)in_sizes; (void)n_in; (void)out_size; (void)ws_size;

    const float* z = (const float*)d_in[0];
    float* out = (float*)d_out;

    char* ws = (char*)d_ws;                         // workspace layout (~50 KB)
    float*    ns       = (float*)(ws);              // 4096 f
    float*    cc       = (float*)(ws + 16384);      // 4096 f
    unsigned* bins     = (unsigned*)(ws + 32768);   // 256 u32
    Ctrl*     ctrl     = (Ctrl*)(ws + 33792);       // 16 B
    float*    partials = (float*)(ws + 34816);      // 4096 f

    norms_kernel<<<kN, kD, 0, stream>>>(z, ns, cc);
    init_kernel<<<1, 256, 0, stream>>>(bins, ctrl);

    const dim3 grid(kN / 64, kN / 64);
    for (int p = 0; p < 4; ++p) {
        hist_kernel<<<grid, 128, 0, stream>>>(z, ns, bins, ctrl, p);
        select_kernel<<<1, 256, 0, stream>>>(bins, ctrl, p);
    }

    reduce_kernel<<<grid, 128, 0, stream>>>(z, ns, cc, ctrl, partials);
    final_kernel<<<1, 256, 0, stream>>>(partials, out);
}